// PillarFeatureNet_38903813767719
// MI455X (gfx1250) — compile-verified
//
#include <hip/hip_runtime.h>
#include <math.h>

typedef __attribute__((ext_vector_type(2))) float v2f;
typedef __attribute__((ext_vector_type(8))) float v8f;

#define NSTAT 54          // 9 feature sums + 45 upper-tri second moments
#define STAT_BLOCKS 120   // blocks for pass-1 partial reduction

// ---------------------------------------------------------------------------
// Pass 1: per-block partial sums of [sum f_j (9), sum f_j*f_k j<=k (45)]
// over all masked points. One wave == one pillar (lane == point index).
// Deterministic: partials written per block, reduced in pass 2.
// ---------------------------------------------------------------------------
__global__ void __launch_bounds__(256)
pfn_stats_kernel(const float* __restrict__ voxels,
                 const int* __restrict__ num_points,
                 const int* __restrict__ coors,
                 float* __restrict__ partials, int npillars) {
  const int lane = threadIdx.x & 31;
  const int wave = threadIdx.x >> 5;
  const int wpb  = blockDim.x >> 5;

  float acc[NSTAT];
#pragma unroll
  for (int i = 0; i < NSTAT; ++i) acc[i] = 0.0f;

  for (int n = blockIdx.x * wpb + wave; n < npillars; n += gridDim.x * wpb) {
    const float4 v = ((const float4*)voxels)[n * 32 + lane];
    const int npts = num_points[n];
    // unmasked xyz sum over all 32 points (matches reference)
    float sx = v.x, sy = v.y, sz = v.z;
#pragma unroll
    for (int off = 16; off; off >>= 1) {
      sx += __shfl_xor(sx, off);
      sy += __shfl_xor(sy, off);
      sz += __shfl_xor(sz, off);
    }
    const float inv = 1.0f / (float)max(npts, 1);
    const float px = (float)coors[n * 4 + 1] * 0.1f + (0.05f - 20.0f);
    const float py = (float)coors[n * 4 + 2] * 0.1f + (0.05f - 20.0f);
    const float m = (lane < npts) ? 1.0f : 0.0f;
    float f[9];
    f[0] = v.x * m; f[1] = v.y * m; f[2] = v.z * m; f[3] = v.w * m;
    f[4] = (v.x - sx * inv) * m;
    f[5] = (v.y - sy * inv) * m;
    f[6] = (v.z - sz * inv) * m;
    f[7] = (v.x - px) * m;
    f[8] = (v.y - py) * m;
#pragma unroll
    for (int j = 0; j < 9; ++j) acc[j] += f[j];
    int idx = 9;
#pragma unroll
    for (int j = 0; j < 9; ++j)
#pragma unroll
      for (int k = j; k < 9; ++k) acc[idx++] += f[j] * f[k];
  }

  // wave reduction
#pragma unroll
  for (int i = 0; i < NSTAT; ++i) {
    float a = acc[i];
#pragma unroll
    for (int off = 16; off; off >>= 1) a += __shfl_xor(a, off);
    acc[i] = a;
  }

  __shared__ float red[8][NSTAT];
  if (lane == 0) {
#pragma unroll
    for (int i = 0; i < NSTAT; ++i) red[wave][i] = acc[i];
  }
  __syncthreads();
  if ((int)threadIdx.x < NSTAT) {
    float t = 0.0f;
    for (int w = 0; w < wpb; ++w) t += red[w][threadIdx.x];
    partials[blockIdx.x * NSTAT + threadIdx.x] = t;
  }
}

// ---------------------------------------------------------------------------
// Pass 2: fold sufficient statistics through W into per-channel BN scale/bias.
//   mean[o]  = (W[o]·sumf)/M
//   E[x2][o] = (W[o]^T S W[o])/M,  var = E[x2] - mean^2
//   scale[o] = gamma[o]*rsqrt(var+eps),  bias[o] = beta[o] - mean*scale
// ---------------------------------------------------------------------------
__global__ void pfn_bn_kernel(const float* __restrict__ partials,
                              const float* __restrict__ W,
                              const float* __restrict__ gamma,
                              const float* __restrict__ beta,
                              float* __restrict__ scale_bias,
                              int nblocks, int npillars) {
  __shared__ float tot[NSTAT];
  if ((int)threadIdx.x < NSTAT) {
    float t = 0.0f;
    for (int b = 0; b < nblocks; ++b) t += partials[b * NSTAT + threadIdx.x];
    tot[threadIdx.x] = t;
  }
  __syncthreads();
  const int o = threadIdx.x;
  if (o < 64) {
    float w[9];
#pragma unroll
    for (int c = 0; c < 9; ++c) w[c] = W[o * 9 + c];
    const float Mf = (float)npillars * 32.0f;
    float s1 = 0.0f;
#pragma unroll
    for (int c = 0; c < 9; ++c) s1 += w[c] * tot[c];
    const float mean = s1 / Mf;
    float s2 = 0.0f;
    int idx = 9;
#pragma unroll
    for (int j = 0; j < 9; ++j)
#pragma unroll
      for (int k = j; k < 9; ++k) {
        const float term = w[j] * w[k] * tot[idx++];
        s2 += (k == j) ? term : 2.0f * term;
      }
    const float var = s2 / Mf - mean * mean;
    const float sc = gamma[o] * rsqrtf(var + 0.001f);
    scale_bias[o] = sc;
    scale_bias[64 + o] = beta[o] - mean * sc;
  }
}

// ---------------------------------------------------------------------------
// Pass 3: per-pillar fused augment + GEMM (WMMA f32 16x16x4, K padded to 12)
// + BN + ReLU + masked max. One wave per pillar; block = 8 waves.
// A lane L: M = L%16, VGPRj -> K = 4*kk + 2*(L/16) + j  (contig float2 in LDS)
// B lane L: N = L%16, VGPRj -> K = 4*kk + 2*(L/16) + j
// C lane L: channel o = 16*t + L%16, VGPR v -> row M = v + 8*(L/16)
//
// Masked max identity: since relu is monotone and cmax starts at 0,
//   max_{valid p} relu(x_p) == max(0, max_p x_p * mask_p),  mask in {0,1}
// which also yields 0 when num_points==0 (matching the reference zero-fill).
// Fully branch-free epilogue: EXEC stays all-ones for every WMMA.
// ---------------------------------------------------------------------------
__global__ void __launch_bounds__(256, 2)
pfn_main_kernel(const float* __restrict__ voxels,
                const int* __restrict__ num_points,
                const int* __restrict__ coors,
                const float* __restrict__ W,
                const float* __restrict__ scale_bias,
                float* __restrict__ out, int npillars) {
  __shared__ float Bt[64][12];        // W transposed + zero pad K=9..11
  __shared__ float feat[8][32][12];   // per-wave 32 points x 12 (padded)

  const int tid  = threadIdx.x;
  const int lane = tid & 31;
  const int wave = tid >> 5;
  const int hi   = lane >> 4;   // half-wave
  const int lo   = lane & 15;

  for (int i = tid; i < 64 * 12; i += blockDim.x) {
    const int o = i / 12, k = i % 12;
    Bt[o][k] = (k < 9) ? W[o * 9 + k] : 0.0f;
  }

  int n = blockIdx.x * 8 + wave;
  const bool valid = (n < npillars);
  if (!valid) n = 0;  // clamp so every wave runs full code (uniform barriers)

  const float4 v = ((const float4*)voxels)[n * 32 + lane];
  const int npts = num_points[n];
  float sx = v.x, sy = v.y, sz = v.z;
#pragma unroll
  for (int off = 16; off; off >>= 1) {
    sx += __shfl_xor(sx, off);
    sy += __shfl_xor(sy, off);
    sz += __shfl_xor(sz, off);
  }
  const float inv = 1.0f / (float)max(npts, 1);
  const float px = (float)coors[n * 4 + 1] * 0.1f + (0.05f - 20.0f);
  const float py = (float)coors[n * 4 + 2] * 0.1f + (0.05f - 20.0f);
  const float m = (lane < npts) ? 1.0f : 0.0f;

  feat[wave][lane][0] = v.x * m;
  feat[wave][lane][1] = v.y * m;
  feat[wave][lane][2] = v.z * m;
  feat[wave][lane][3] = v.w * m;
  feat[wave][lane][4] = (v.x - sx * inv) * m;
  feat[wave][lane][5] = (v.y - sy * inv) * m;
  feat[wave][lane][6] = (v.z - sz * inv) * m;
  feat[wave][lane][7] = (v.x - px) * m;
  feat[wave][lane][8] = (v.y - py) * m;
  feat[wave][lane][9] = 0.0f;
  feat[wave][lane][10] = 0.0f;
  feat[wave][lane][11] = 0.0f;

  __syncthreads();

  // B fragments (pillar independent): 4 channel tiles x 3 K-steps
  v2f Bf[4][3];
#pragma unroll
  for (int t = 0; t < 4; ++t)
#pragma unroll
    for (int kk = 0; kk < 3; ++kk)
      Bf[t][kk] = *(const v2f*)&Bt[16 * t + lo][4 * kk + 2 * hi];

  float sc[4], bi[4];
#pragma unroll
  for (int t = 0; t < 4; ++t) {
    sc[t] = scale_bias[16 * t + lo];
    bi[t] = scale_bias[64 + 16 * t + lo];
  }

  // per-row {0,1} validity masks, hoisted out of all compute loops
  float pm[2][8];
#pragma unroll
  for (int mt = 0; mt < 2; ++mt)
#pragma unroll
    for (int vv = 0; vv < 8; ++vv)
      pm[mt][vv] = (16 * mt + vv + 8 * hi < npts) ? 1.0f : 0.0f;

  float cmax[4] = {0.0f, 0.0f, 0.0f, 0.0f};

#pragma unroll
  for (int mt = 0; mt < 2; ++mt) {
    v2f Af[3];
#pragma unroll
    for (int kk = 0; kk < 3; ++kk)
      Af[kk] = *(const v2f*)&feat[wave][16 * mt + lo][4 * kk + 2 * hi];

#pragma unroll
    for (int t = 0; t < 4; ++t) {
      v8f c = {};
#pragma unroll
      for (int kk = 0; kk < 3; ++kk)
        c = __builtin_amdgcn_wmma_f32_16x16x4_f32(
            false, Af[kk], false, Bf[t][kk], (short)0, c, false, false);
      // branch-free BN + masked max (relu folded into cmax>=0 invariant)
#pragma unroll
      for (int vv = 0; vv < 8; ++vv) {
        const float x = fmaf(c[vv], sc[t], bi[t]);
        cmax[t] = fmaxf(cmax[t], x * pm[mt][vv]);
      }
    }
  }

  // combine the two half-wave row groups (same channel in lane L and L^16)
#pragma unroll
  for (int t = 0; t < 4; ++t)
    cmax[t] = fmaxf(cmax[t], __shfl_xor(cmax[t], 16));

  if (valid && lane < 16) {
#pragma unroll
    for (int t = 0; t < 4; ++t)
      out[n * 64 + 16 * t + lo] = cmax[t];
  }
}

// ---------------------------------------------------------------------------
extern "C" void kernel_launch(void* const* d_in, const int* in_sizes, int n_in,
                              void* d_out, int out_size, void* d_ws, size_t ws_size,
                              hipStream_t stream) {
  const float* voxels     = (const float*)d_in[0];
  const int*   num_points = (const int*)d_in[1];
  const int*   coors      = (const int*)d_in[2];
  const float* W          = (const float*)d_in[3];
  const float* gamma      = (const float*)d_in[4];
  const float* beta       = (const float*)d_in[5];
  float* out = (float*)d_out;

  const int npillars = in_sizes[1];

  float* partials   = (float*)d_ws;                    // STAT_BLOCKS * 54
  float* scale_bias = partials + STAT_BLOCKS * NSTAT;  // 128 floats

  pfn_stats_kernel<<<STAT_BLOCKS, 256, 0, stream>>>(
      voxels, num_points, coors, partials, npillars);

  pfn_bn_kernel<<<1, 64, 0, stream>>>(
      partials, W, gamma, beta, scale_bias, STAT_BLOCKS, npillars);

  const int blocks = (npillars + 7) / 8;  // 8 pillars (waves) per block
  pfn_main_kernel<<<blocks, 256, 0, stream>>>(
      voxels, num_points, coors, W, scale_bias, out, npillars);
}